// ROIAlign_26087631356447
// MI455X (gfx1250) — compile-verified
//
#include <hip/hip_runtime.h>
#include <math.h>

// ---------------------------------------------------------------------------
// Multi-level ROIAlign for MI455X (gfx1250, wave32).
// Inputs (setup_inputs order):
//   d_in[0..3] : p2..p5  float32 NCHW, B=4, C=256, dims 200/100/50/25
//   d_in[4]    : proposals float32 [B,K,4] = [4,512,4]
//   d_in[5]    : H (int, 1 elem)   d_in[6] : W (int, 1 elem)
// Output: [N=2048, C=256, 7, 7] float32
//
// Structure: block = one ROI x 32 channels; lane = sample point (wave32-aware:
// each output row's 7 lanes share the same two feature rows -> high cacheline
// utilization on the gather; 49-contiguous output floats per (n,c) coalesce).
// Box staged via CDNA5 async global->LDS DMA; per-plane corner pairs loaded
// as single b64; output written with non-temporal stores to protect L2.
// ---------------------------------------------------------------------------

#define OUT_SIZE      7
#define NPTS          49          // 7*7
#define C_TOTAL       256
#define K_PER_B       512         // proposals per batch image
#define CH_PER_BLOCK  32
#define CH_PER_THREAD 8           // CH_PER_BLOCK / blockDim.y

#if defined(__gfx1250__) && __has_builtin(__builtin_amdgcn_global_load_async_to_lds_b32)
#define HAVE_ASYNC_LDS 1
#else
#define HAVE_ASYNC_LDS 0
#endif

typedef __attribute__((address_space(1))) int gint_t;   // global int
typedef __attribute__((address_space(3))) int lint_t;   // LDS int

// 8-byte vector with only 4-byte alignment guarantee -> still lowers to
// global_load_b64 (gfx1250 tolerates dword-aligned b64).
typedef float f2_t __attribute__((ext_vector_type(2), aligned(4)));

__global__ __launch_bounds__(256) void ROIAlign_26087631356447_kernel(
    const float* __restrict__ p2, const float* __restrict__ p3,
    const float* __restrict__ p4, const float* __restrict__ p5,
    const float* __restrict__ prop,
    const int* __restrict__ Hp, const int* __restrict__ Wp,
    float* __restrict__ out)
{
    const int n   = blockIdx.x;                    // ROI index, 0..2047
    const int tid = threadIdx.y * 64 + threadIdx.x;

    __shared__ float s_box[4];
    __shared__ float s_wx[OUT_SIZE], s_wy[OUT_SIZE];
    __shared__ int   s_xl[OUT_SIZE];               // left index of x corner pair
    __shared__ int   s_y0[OUT_SIZE], s_y1[OUT_SIZE];

    // ---- Stage the ROI box into LDS via CDNA5 async DMA --------------------
#if HAVE_ASYNC_LDS
    if (tid < 4) {
        __builtin_amdgcn_global_load_async_to_lds_b32(
            (gint_t*)(prop + (size_t)4 * n + tid),
            (lint_t*)&s_box[tid],
            /*offset=*/0, /*cpol=*/0);
    }
#if __has_builtin(__builtin_amdgcn_s_wait_asynccnt)
    __builtin_amdgcn_s_wait_asynccnt(0);
#else
    asm volatile("s_wait_asynccnt 0" ::: "memory");
#endif
#else
    if (tid < 4) s_box[tid] = prop[(size_t)4 * n + tid];
#endif
    __syncthreads();

    const float bx1 = s_box[0], by1 = s_box[1], bx2 = s_box[2], by2 = s_box[3];

    // ---- FPN level selection (uniform per block -> SALU after readfirstlane)
    const float area = fmaxf((bx2 - bx1) * (by2 - by1), 1e-6f);
    const float side = __builtin_sqrtf(area);
    int lvl = (int)floorf(4.0f + __builtin_log2f(side * (1.0f / 224.0f)));
    lvl = min(max(lvl, 2), 5) - 2;                 // 0..3  -> p2..p5

    const int Wf = (lvl == 0) ? 200 : (lvl == 1) ? 100 : (lvl == 2) ? 50 : 25;
    const int Hf = Wf;                             // all levels square here
    const float* __restrict__ fmp =
        (lvl == 0) ? p2 : (lvl == 1) ? p3 : (lvl == 2) ? p4 : p5;

    const float Wim = (float)Wp[0];
    const float Him = (float)Hp[0];

    // ---- 7 x-coords (wave 0) and 7 y-coords (wave 2), broadcast via LDS ----
    if (threadIdx.y == 0 && threadIdx.x < OUT_SIZE) {
        const float g  = (float)threadIdx.x * (1.0f / 6.0f);
        float xc = (bx1 + (bx2 - bx1) * g) * (1.0f / Wim) * (float)(Wf - 1);
        xc = fminf(fmaxf(xc, 0.0f), (float)(Wf - 1));
        const int x0 = (int)floorf(xc);
        const int xl = min(x0, Wf - 2);            // pair [xl, xl+1] always valid
        s_xl[threadIdx.x] = xl;
        // adjusted weight: value = f[xl] + wx' * (f[xl+1] - f[xl]);
        // at the right border x0==Wf-1 -> wx'=1 selects f[Wf-1] exactly.
        s_wx[threadIdx.x] = (xc - (float)x0) + (float)(x0 - xl);
    }
    if (threadIdx.y == 1 && threadIdx.x < OUT_SIZE) {
        const float g  = (float)threadIdx.x * (1.0f / 6.0f);
        float yc = (by1 + (by2 - by1) * g) * (1.0f / Him) * (float)(Hf - 1);
        yc = fminf(fmaxf(yc, 0.0f), (float)(Hf - 1));
        const int y0 = (int)floorf(yc);
        s_y0[threadIdx.x] = y0;
        s_y1[threadIdx.x] = min(y0 + 1, Hf - 1);
        s_wy[threadIdx.x] = yc - (float)y0;
    }
    __syncthreads();

    // ---- Gather (one b64 per row) + bilinear combine -----------------------
    const int p = threadIdx.x;
    if (p < NPTS) {
        const int px = p % OUT_SIZE;
        const int py = p / OUT_SIZE;

        const int   xl = s_xl[px];
        const int   y0 = s_y0[py], y1 = s_y1[py];
        const float wx = s_wx[px], wy = s_wy[py];

        const size_t HfWf  = (size_t)Hf * (size_t)Wf;
        const int    b     = n >> 9;               // n / K_PER_B
        const int    cbase = blockIdx.y * CH_PER_BLOCK + threadIdx.y * CH_PER_THREAD;

        const float* __restrict__ plane =
            fmp + (size_t)(b * C_TOTAL + cbase) * HfWf;
        const size_t r0 = (size_t)y0 * (size_t)Wf + (size_t)xl;
        const size_t r1 = (size_t)y1 * (size_t)Wf + (size_t)xl;

        float* __restrict__ op =
            out + ((size_t)n * C_TOTAL + (size_t)cbase) * NPTS + p;

#pragma unroll
        for (int j = 0; j < CH_PER_THREAD; ++j) {
            const float* __restrict__ a0 = plane + r0;
            const float* __restrict__ a1 = plane + r1;

            if (j + 1 < CH_PER_THREAD) {
                // next channel plane is HfWf*4 bytes away: software prefetch
                __builtin_prefetch(a0 + HfWf, 0, 3);
                __builtin_prefetch(a1 + HfWf, 0, 3);
            }

            const f2_t pr0 = *(const f2_t*)a0;     // f[y0][xl], f[y0][xl+1]
            const f2_t pr1 = *(const f2_t*)a1;     // f[y1][xl], f[y1][xl+1]

            const float top = pr0.x + wx * (pr0.y - pr0.x);
            const float bot = pr1.x + wx * (pr1.y - pr1.x);
            const float r   = top + wy * (bot - top);

            // streaming store: keep the 192MB L2 for feature-map lines
            __builtin_nontemporal_store(r, op);

            plane += HfWf;
            op    += NPTS;
        }
    }
}

extern "C" void kernel_launch(void* const* d_in, const int* in_sizes, int n_in,
                              void* d_out, int out_size, void* d_ws, size_t ws_size,
                              hipStream_t stream) {
    (void)n_in; (void)out_size; (void)d_ws; (void)ws_size;

    const float* p2   = (const float*)d_in[0];
    const float* p3   = (const float*)d_in[1];
    const float* p4   = (const float*)d_in[2];
    const float* p5   = (const float*)d_in[3];
    const float* prop = (const float*)d_in[4];
    const int*   Hp   = (const int*)d_in[5];
    const int*   Wp   = (const int*)d_in[6];

    const int N = in_sizes[4] / 4;                 // 2048 ROIs

    dim3 grid(N, C_TOTAL / CH_PER_BLOCK);          // (2048, 8)
    dim3 block(64, 4);                             // 8 wave32 per block

    hipLaunchKernelGGL(ROIAlign_26087631356447_kernel, grid, block, 0, stream,
                       p2, p3, p4, p5, prop, Hp, Wp, (float*)d_out);
}